// EquiMessagePsuedo_34376918237208
// MI455X (gfx1250) — compile-verified
//
#include <hip/hip_runtime.h>
#include <hip/hip_bf16.h>

typedef float v2f __attribute__((ext_vector_type(2)));
typedef float v8f __attribute__((ext_vector_type(8)));
typedef unsigned int u32;
typedef u32 u32x4 __attribute__((ext_vector_type(4)));
typedef int  i32x4 __attribute__((ext_vector_type(4)));
typedef int  i32x8 __attribute__((ext_vector_type(8)));

#define NNODES 10000
#define FDIM   64
#define NEDGES 320000
#define NRBF   20
#define CDIM   576          // 9*F
#define PI_OVER_CUT 0.6283185307179586f
#define CUTOFF 5.0f
#define WAVES  8

__device__ __forceinline__ v8f wmma_f32_k4(v2f a, v2f b, v8f c) {
    // D = A(16x4 f32) * B(4x16 f32) + C(16x16 f32)
    return __builtin_amdgcn_wmma_f32_16x16x4_f32(
        /*neg_a=*/false, a, /*neg_b=*/false, b,
        /*c_mod=*/(short)0, c, /*reuse_a=*/false, /*reuse_b=*/false);
}

__global__ void zero_out_kernel(float* out, long n) {
    long idx = (long)blockIdx.x * blockDim.x + threadIdx.x;
    if (idx < n) out[idx] = 0.0f;
}

// One wave (32 threads) per 16 nodes: phi = silu(s@W1+b1) @ W2 + b2
__global__ void node_phi_kernel(const float* __restrict__ s_j,
                                const float* __restrict__ W1,
                                const float* __restrict__ b1,
                                const float* __restrict__ W2,
                                const float* __restrict__ b2,
                                float* __restrict__ phi) {
    __shared__ float hS[16][FDIM];     // silu(s@W1+b1) tile, 4 KB
    const int lane = threadIdx.x & 31;
    const int lo   = lane & 15;
    const int hi   = lane >> 4;
    const long nb  = (long)blockIdx.x * 16;   // 625 blocks * 16 = 10000 exactly

    // ---- h = silu(s @ W1 + b1), 16x64 output, 4 column tiles ----
    for (int ct = 0; ct < 4; ++ct) {
        const int cb = ct * 16;
        const float bias = b1[cb + lo];
        v8f acc = {bias, bias, bias, bias, bias, bias, bias, bias};
#pragma unroll
        for (int kk = 0; kk < FDIM; kk += 4) {
            const int k0 = kk + 2 * hi;
            v2f a, b;
            a.x = s_j[(nb + lo) * FDIM + k0];
            a.y = s_j[(nb + lo) * FDIM + k0 + 1];
            b.x = W1[(long)k0 * FDIM + cb + lo];
            b.y = W1[(long)(k0 + 1) * FDIM + cb + lo];
            acc = wmma_f32_k4(a, b, acc);
        }
#pragma unroll
        for (int r = 0; r < 8; ++r) {
            const int M = r + hi * 8;
            const float x = acc[r];
            hS[M][cb + lo] = __fdividef(x, 1.0f + __expf(-x));   // fast silu: v_rcp+v_mul
        }
    }
    __syncthreads();   // single-wave workgroup: lowers to dscnt wait

    // ---- phi = h @ W2 + b2, 16x576 output, 36 column tiles ----
    for (int t = 0; t < 36; ++t) {
        const int cb = t * 16;
        const float bias = b2[cb + lo];
        v8f acc = {bias, bias, bias, bias, bias, bias, bias, bias};
#pragma unroll
        for (int kk = 0; kk < FDIM; kk += 4) {
            const int k0 = kk + 2 * hi;
            v2f a, b;
            a.x = hS[lo][k0];
            a.y = hS[lo][k0 + 1];
            b.x = W2[(long)k0 * CDIM + cb + lo];
            b.y = W2[(long)(k0 + 1) * CDIM + cb + lo];
            acc = wmma_f32_k4(a, b, acc);
        }
#pragma unroll
        for (int r = 0; r < 8; ++r) {
            const int M = r + hi * 8;
            phi[(nb + M) * CDIM + cb + lo] = acc[r];
        }
    }
}

// 8 waves per block, each wave owns 16 edges. 2500 blocks * 8 * 16 = 320000 exactly.
__global__ void __launch_bounds__(256)
edge_kernel(const float* __restrict__ s_j,
            const float* __restrict__ sbar_j,
            const float* __restrict__ v_j,
            const float* __restrict__ vbar_j,
            const float* __restrict__ r_ij,
            const long long* __restrict__ nbrs,
            const float* __restrict__ W_rbf,
            const float* __restrict__ b_rbf,
            const float* __restrict__ phi,
            float* __restrict__ dh,
            float* __restrict__ dhbar,
            float* __restrict__ dv,
            float* __restrict__ dvbar) {
    __shared__ float WldsS[NRBF][CDIM];      // 45 KB: W_rbf staged via TDM
    __shared__ float rbfS[WAVES][16][NRBF];
    __shared__ float envS[WAVES][16];
    __shared__ float unitS[WAVES][16][3];
    __shared__ int   iS[WAVES][16];
    __shared__ int   jS[WAVES][16];

    const int w    = threadIdx.x >> 5;
    const int lane = threadIdx.x & 31;
    const int lo   = lane & 15;
    const int hi   = lane >> 4;
    const long ebase = ((long)blockIdx.x * WAVES + w) * 16;

#if __has_builtin(__builtin_amdgcn_tensor_load_to_lds)
    // ---- stage W_rbf (20 x 576 f32) into LDS with the Tensor Data Mover ----
    if (w == 0) {
        const u32 lds_off = (u32)(unsigned long long)(&WldsS[0][0]);  // flat addr low 32 = LDS offset
        const unsigned long long ga = (unsigned long long)W_rbf;
        u32x4 g0;
        g0.x = 1u;                                    // count=1, user mode, no gather
        g0.y = lds_off;                               // lds_addr
        g0.z = (u32)ga;                               // global_addr[31:0]
        g0.w = (u32)((ga >> 32) & 0x1FFFFFFu) | (2u << 30);  // global_addr[56:32] | type=2
        i32x8 g1;
        g1[0] = (int)(2u << 16);                      // wg_mask=0, data_size=2 (4 bytes)
        g1[1] = (int)((u32)CDIM << 16);               // tensor_dim0[15:0] = 576
        g1[2] = (int)((u32)NRBF << 16);               // dim0 hi=0 | tensor_dim1[15:0] = 20
        g1[3] = (int)((u32)CDIM << 16);               // dim1 hi=0 | tile_dim0 = 576
        g1[4] = (int)NRBF;                            // tile_dim1 = 20, tile_dim2 = 0
        g1[5] = (int)CDIM;                            // tensor_dim0_stride[31:0] = 576
        g1[6] = 0;                                    // stride0 hi | stride1 lo
        g1[7] = 0;                                    // stride1 hi
        i32x4 gz4 = {0, 0, 0, 0};
        i32x8 gz8 = {0, 0, 0, 0, 0, 0, 0, 0};
        // 6-arg form (this toolchain): (g0, g1, g2, g3, g4, cpol)
        __builtin_amdgcn_tensor_load_to_lds(g0, g1, gz4, gz4, gz8, 0);
        __builtin_amdgcn_s_wait_tensorcnt(0);
    }
#else
    for (int idx = threadIdx.x; idx < NRBF * CDIM; idx += 256)
        (&WldsS[0][0])[idx] = W_rbf[idx];
#endif

    // ---- per-edge scalars: dist, unit, env, 20 RBFs via sin(n*theta) recurrence ----
    if (lane < 16) {
        const long e = ebase + lane;
        const float x = r_ij[e * 3 + 0];
        const float y = r_ij[e * 3 + 1];
        const float z = r_ij[e * 3 + 2];
        const float dist = sqrtf(x * x + y * y + z * z + 3e-8f);
        const float inv_d = 1.0f / dist;
        unitS[w][lane][0] = x * inv_d;
        unitS[w][lane][1] = y * inv_d;
        unitS[w][lane][2] = z * inv_d;
        float sn, cs;
        __sincosf(PI_OVER_CUT * dist, &sn, &cs);
        envS[w][lane] = (dist < CUTOFF) ? 0.5f * (cs + 1.0f) : 0.0f;
        // sin(n*theta): s_{n+1} = 2cos(theta)*s_n - s_{n-1}
        const float twoc = 2.0f * cs;
        float sm1 = 0.0f, s0 = sn;
#pragma unroll
        for (int n = 0; n < NRBF; ++n) {
            rbfS[w][lane][n] = s0 * inv_d;
            const float s1 = twoc * s0 - sm1;
            sm1 = s0; s0 = s1;
        }
        iS[w][lane] = (int)nbrs[e * 2 + 0];
        jS[w][lane] = (int)nbrs[e * 2 + 1];
    }
    __syncthreads();

    // ---- d_sbar_ij = sum_d v_j[i]*vbar_j[j] -> dhbar[i] ----
    for (int e2 = 0; e2 < 16; ++e2) {
        const long i = iS[w][e2];
        const long j = jS[w][e2];
        for (int f = lane; f < FDIM; f += 32) {
            const float* vi  = v_j   + (i * FDIM + f) * 3;
            const float* vbj = vbar_j + (j * FDIM + f) * 3;
            const float dot = vi[0] * vbj[0] + vi[1] * vbj[1] + vi[2] * vbj[2];
            atomicAdd(&dhbar[i * FDIM + f], dot);
        }
    }

    // ---- w_s = rbf @ W_rbf + b_rbf (WMMA, K=20 = 5 steps), fused with scatter ----
    for (int t = 0; t < 36; ++t) {
        const int g     = t >> 2;          // group 0..8 (uniform per tile)
        const int fbase = (t & 3) * 16;    // feature base within group
        const int cb    = t * 16;          // column in 576
        const float bias = b_rbf[cb + lo];
        v8f acc = {bias, bias, bias, bias, bias, bias, bias, bias};
#pragma unroll
        for (int kk = 0; kk < NRBF; kk += 4) {
            const int k0 = kk + 2 * hi;
            v2f a, b;
            a.x = rbfS[w][lo][k0];
            a.y = rbfS[w][lo][k0 + 1];
            b.x = WldsS[k0][cb + lo];
            b.y = WldsS[k0 + 1][cb + lo];
            acc = wmma_f32_k4(a, b, acc);
        }
        const int f = fbase + lo;
#pragma unroll
        for (int r = 0; r < 8; ++r) {
            const int M = r + hi * 8;            // edge row within tile
            const float ws  = acc[r] * envS[w][M];
            const long  i   = iS[w][M];
            const long  j   = jS[w][M];
            const float inv = phi[j * CDIM + cb + lo] * ws;
            const long ib = i * FDIM + f;
            const long jb = j * FDIM + f;
            if (g == 0) {                                   // s0 * s_j[i] -> dh
                atomicAdd(&dh[ib], inv * s_j[ib]);
            } else if (g == 1) {                            // sp0 * unit -> dv
                atomicAdd(&dv[ib * 3 + 0], inv * unitS[w][M][0]);
                atomicAdd(&dv[ib * 3 + 1], inv * unitS[w][M][1]);
                atomicAdd(&dv[ib * 3 + 2], inv * unitS[w][M][2]);
            } else if (g == 2) {                            // sp1 * v_j[j] -> dv
                const float* vj = v_j + jb * 3;
                atomicAdd(&dv[ib * 3 + 0], inv * vj[0]);
                atomicAdd(&dv[ib * 3 + 1], inv * vj[1]);
                atomicAdd(&dv[ib * 3 + 2], inv * vj[2]);
            } else if (g == 3) {                            // sp2 * cross(vi, vbj) -> dv
                const float* vi  = v_j    + ib * 3;
                const float* vbj = vbar_j + jb * 3;
                atomicAdd(&dv[ib * 3 + 0], inv * (vi[1] * vbj[2] - vi[2] * vbj[1]));
                atomicAdd(&dv[ib * 3 + 1], inv * (vi[2] * vbj[0] - vi[0] * vbj[2]));
                atomicAdd(&dv[ib * 3 + 2], inv * (vi[0] * vbj[1] - vi[1] * vbj[0]));
            } else if (g == 4) {                            // sp3 * sbar_i * vbj -> dv
                const float s  = inv * sbar_j[ib];
                const float* vbj = vbar_j + jb * 3;
                atomicAdd(&dv[ib * 3 + 0], s * vbj[0]);
                atomicAdd(&dv[ib * 3 + 1], s * vbj[1]);
                atomicAdd(&dv[ib * 3 + 2], s * vbj[2]);
            } else if (g == 5) {                            // sp4 * vbj -> dvbar
                const float* vbj = vbar_j + jb * 3;
                atomicAdd(&dvbar[ib * 3 + 0], inv * vbj[0]);
                atomicAdd(&dvbar[ib * 3 + 1], inv * vbj[1]);
                atomicAdd(&dvbar[ib * 3 + 2], inv * vbj[2]);
            } else if (g == 6) {                            // sp5 * sbar_i * vj -> dvbar
                const float s  = inv * sbar_j[ib];
                const float* vj = v_j + jb * 3;
                atomicAdd(&dvbar[ib * 3 + 0], s * vj[0]);
                atomicAdd(&dvbar[ib * 3 + 1], s * vj[1]);
                atomicAdd(&dvbar[ib * 3 + 2], s * vj[2]);
            } else if (g == 7) {                            // sp6 * cross(vi, vj) -> dvbar
                const float* vi = v_j + ib * 3;
                const float* vj = v_j + jb * 3;
                atomicAdd(&dvbar[ib * 3 + 0], inv * (vi[1] * vj[2] - vi[2] * vj[1]));
                atomicAdd(&dvbar[ib * 3 + 1], inv * (vi[2] * vj[0] - vi[0] * vj[2]));
                atomicAdd(&dvbar[ib * 3 + 2], inv * (vi[0] * vj[1] - vi[1] * vj[0]));
            } else {                                        // sp7 * cross(vbi, vbj) -> dvbar
                const float* vbi = vbar_j + ib * 3;
                const float* vbj = vbar_j + jb * 3;
                atomicAdd(&dvbar[ib * 3 + 0], inv * (vbi[1] * vbj[2] - vbi[2] * vbj[1]));
                atomicAdd(&dvbar[ib * 3 + 1], inv * (vbi[2] * vbj[0] - vbi[0] * vbj[2]));
                atomicAdd(&dvbar[ib * 3 + 2], inv * (vbi[0] * vbj[1] - vbi[1] * vbj[0]));
            }
        }
    }
}

extern "C" void kernel_launch(void* const* d_in, const int* in_sizes, int n_in,
                              void* d_out, int out_size, void* d_ws, size_t ws_size,
                              hipStream_t stream) {
    const float*     s_j    = (const float*)d_in[0];
    const float*     sbar_j = (const float*)d_in[1];
    const float*     v_j    = (const float*)d_in[2];
    const float*     vbar_j = (const float*)d_in[3];
    const float*     r_ij   = (const float*)d_in[4];
    const long long* nbrs   = (const long long*)d_in[5];
    const float*     W1     = (const float*)d_in[6];
    const float*     b1     = (const float*)d_in[7];
    const float*     W2     = (const float*)d_in[8];
    const float*     b2     = (const float*)d_in[9];
    const float*     W_rbf  = (const float*)d_in[10];
    const float*     b_rbf  = (const float*)d_in[11];

    float* out   = (float*)d_out;
    float* dh    = out;                                        // N*F
    float* dhbar = out + (long)NNODES * FDIM;                  // N*F
    float* dv    = out + 2L * NNODES * FDIM;                   // N*F*3
    float* dvbar = out + 2L * NNODES * FDIM + 3L * NNODES * FDIM;

    float* phi = (float*)d_ws;                                 // N*576 f32 = 23 MB

    const long total_out = 2L * NNODES * FDIM + 6L * NNODES * FDIM;  // 5,120,000
    zero_out_kernel<<<(int)((total_out + 255) / 256), 256, 0, stream>>>(out, total_out);

    node_phi_kernel<<<NNODES / 16, 32, 0, stream>>>(s_j, W1, b1, W2, b2, phi);

    edge_kernel<<<NEDGES / (16 * WAVES), 32 * WAVES, 0, stream>>>(
        s_j, sbar_j, v_j, vbar_j, r_ij, nbrs, W_rbf, b_rbf, phi,
        dh, dhbar, dv, dvbar);
}